// SSM_20684562497901
// MI455X (gfx1250) — compile-verified
//
#include <hip/hip_runtime.h>
#include <hip/hip_bf16.h>

// ---------------------------------------------------------------------------
// Mamba selective scan for MI455X (gfx1250, wave32).
//   B=2, L=2048, D_INNER=2048, DT_RANK=64, D_STATE=16
// Pipeline:
//   K1: x_dbl[M=4096,96]  = x[M,2048] @ x_proj_w[96,2048]^T      (f32 WMMA 16x16x4)
//   K2: delta[M,2048]     = softplus(x_dbl[:, :64] @ dt_proj_w^T + b)  (f32 WMMA)
//   K3: fused selective scan; B/C tiles double-buffered into LDS with
//       GLOBAL_LOAD_ASYNC_TO_LDS_B128 (ASYNCcnt), h[16] per (b,d) thread.
// ---------------------------------------------------------------------------

typedef __attribute__((ext_vector_type(2))) float v2f;
typedef __attribute__((ext_vector_type(8))) float v8f;

#define D_INNER 2048
#define DT_RANK 64
#define D_STATE 16
#define BATCH   2
#define SEQLEN  2048
#define M_TOT   (BATCH * SEQLEN)        // 4096 rows (b,l) flattened
#define K_PROJ  (DT_RANK + 2 * D_STATE) // 96

// ===========================================================================
// Kernel 1: x_dbl = x @ x_proj_w^T      (M=4096, K=2048, N=96)
// One wave computes one 16x16 tile with V_WMMA_F32_16X16X4_F32.
// ===========================================================================
__global__ __launch_bounds__(32) void ssm_xproj_wmma(
    const float* __restrict__ X,     // [M_TOT, D_INNER]
    const float* __restrict__ W,     // [K_PROJ, D_INNER]
    float* __restrict__ xdbl)        // [M_TOT, K_PROJ]
{
    const int m0   = blockIdx.x << 4;
    const int n0   = blockIdx.y << 4;
    const int lane = threadIdx.x;
    const int r    = lane & 15;       // A row (M) and B col (N)
    const int kh   = lane >> 4;       // K sub-pair select

    const float* __restrict__ xrow = X + (size_t)(m0 + r) * D_INNER + 2 * kh;
    const float* __restrict__ wrow = W + (size_t)(n0 + r) * D_INNER + 2 * kh;

    v8f c = {};
#pragma unroll 4
    for (int k = 0; k < D_INNER; k += 4) {
        v2f a = *(const v2f*)(xrow + k);
        v2f b = *(const v2f*)(wrow + k);
        c = __builtin_amdgcn_wmma_f32_16x16x4_f32(
                /*neg_a=*/false, a, /*neg_b=*/false, b,
                /*c_mod=*/(short)0, c, /*reuse_a=*/false, /*reuse_b=*/false);
    }
#pragma unroll
    for (int j = 0; j < 8; ++j)
        xdbl[(size_t)(m0 + j + 8 * kh) * K_PROJ + n0 + r] = c[j];
}

// ===========================================================================
// Kernel 2: delta = softplus(x_dbl[:, 0:64] @ dt_proj_w^T + bias)
//           (M=4096, K=64, N=2048)
// ===========================================================================
__device__ __forceinline__ float softplus_f(float v) {
    return (v > 20.0f) ? v : log1pf(__expf(v));
}

__global__ __launch_bounds__(32) void ssm_dtproj_wmma(
    const float* __restrict__ xdbl,  // [M_TOT, K_PROJ] (use cols 0..63)
    const float* __restrict__ W2,    // [D_INNER, DT_RANK]
    const float* __restrict__ bias,  // [D_INNER]
    float* __restrict__ delta)       // [M_TOT, D_INNER]
{
    const int m0   = blockIdx.x << 4;
    const int n0   = blockIdx.y << 4;
    const int lane = threadIdx.x;
    const int r    = lane & 15;
    const int kh   = lane >> 4;

    const float* __restrict__ arow = xdbl + (size_t)(m0 + r) * K_PROJ + 2 * kh;
    const float* __restrict__ wrow = W2   + (size_t)(n0 + r) * DT_RANK + 2 * kh;

    v8f c = {};
#pragma unroll
    for (int k = 0; k < DT_RANK; k += 4) {
        v2f a = *(const v2f*)(arow + k);
        v2f b = *(const v2f*)(wrow + k);
        c = __builtin_amdgcn_wmma_f32_16x16x4_f32(
                false, a, false, b, (short)0, c, false, false);
    }
    const float bv = bias[n0 + r];
#pragma unroll
    for (int j = 0; j < 8; ++j) {
        float sp = softplus_f(c[j] + bv);
        delta[(size_t)(m0 + j + 8 * kh) * D_INNER + n0 + r] = sp;
    }
}

// ===========================================================================
// Kernel 3: fused selective scan with async (memory -> LDS) double buffering.
// grid = (D_INNER/256, BATCH), block = 256 (8 waves). Thread t owns channel
// d = bx*256+t of batch by; keeps h[16] in VGPRs.
// B/C of each timestep are cols 64..95 of the 96-float x_dbl row: one
// contiguous 128-byte run = 8 x b128 chunks, streamed straight into LDS with
// GLOBAL_LOAD_ASYNC_TO_LDS_B128 (no VGPR data, tracked by ASYNCcnt).
// ===========================================================================
#define TL 64   // timesteps per LDS tile

// Issue one async 16-byte global->LDS copy. lds_off is a byte offset within
// the workgroup's LDS allocation (VDST is relative to the wave's LDS base).
__device__ __forceinline__ void async_copy_b128(unsigned lds_off, const float* gaddr) {
    asm volatile("global_load_async_to_lds_b128 %0, %1, off"
                 :: "v"(lds_off), "v"(gaddr)
                 : "memory");
}
__device__ __forceinline__ void wait_async0() {
    asm volatile("s_wait_asynccnt 0x0" ::: "memory");
}

__global__ __launch_bounds__(256) void ssm_scan(
    const float* __restrict__ X,      // [B, L, D]
    const float* __restrict__ xdbl,   // [B*L, 96]  (B at col 64, C at col 80)
    const float* __restrict__ delta,  // [B*L, D]
    const float* __restrict__ A_log,  // [D, 16]
    const float* __restrict__ Dvec,   // [D]
    float* __restrict__ Y)            // [B, L, D]
{
    const int t = threadIdx.x;
    const int d = blockIdx.x * 256 + t;
    const int b = blockIdx.y;

    // BC[buf][s][0..15] = B(t=s), BC[buf][s][16..31] = C(t=s).  16 KB total.
    __shared__ float BC[2][TL][2 * D_STATE];
    const unsigned lds_base = (unsigned)(unsigned long long)(const void*)&BC[0][0][0];

    float An[D_STATE];
#pragma unroll
    for (int n = 0; n < D_STATE; ++n)
        An[n] = -__expf(A_log[(size_t)d * D_STATE + n]);

    float h[D_STATE];
#pragma unroll
    for (int n = 0; n < D_STATE; ++n) h[n] = 0.0f;

    const float Dv = Dvec[d];

    const size_t chan = (size_t)b * SEQLEN * D_INNER + d;
    const float* __restrict__ dptr = delta + chan;
    const float* __restrict__ xptr = X + chan;
    float* __restrict__ yptr       = Y + chan;
    const float* __restrict__ bc   = xdbl + (size_t)b * SEQLEN * K_PROJ;

    // Stage tile starting at l0 into buffer `buf`: 64 rows x 32 floats
    // = 512 b128 chunks; 256 threads -> 2 chunks per thread.
    auto stage_tile = [&](int l0, int buf) {
#pragma unroll
        for (int c = 0; c < 2; ++c) {
            const int i = t + c * 256;          // chunk id 0..511
            const int s = i >> 3;               // timestep within tile
            const int q = i & 7;                // 16B chunk within 128B run
            const float* g = bc + (size_t)(l0 + s) * K_PROJ + DT_RANK + q * 4;
            const unsigned off = lds_base +
                (unsigned)(((buf * TL + s) * (2 * D_STATE) + q * 4) * sizeof(float));
            async_copy_b128(off, g);
        }
    };

    // Prologue: stage tile 0, wait, barrier.
    stage_tile(0, 0);
    wait_async0();
    __syncthreads();

    int buf = 0;
    for (int l0 = 0; l0 < SEQLEN; l0 += TL) {
        // Kick off next tile's async loads into the other buffer.
        if (l0 + TL < SEQLEN) {
            stage_tile(l0 + TL, buf ^ 1);
            __builtin_prefetch(dptr + (size_t)(l0 + TL) * D_INNER, 0, 0);
            __builtin_prefetch(xptr + (size_t)(l0 + TL) * D_INNER, 0, 0);
        }

        // Compute current tile.
        for (int s = 0; s < TL; ++s) {
            const size_t off = (size_t)(l0 + s) * D_INNER;
            const float dlt = dptr[off];
            const float xv  = xptr[off];
            const float dx  = dlt * xv;
            const float* __restrict__ bcs = &BC[buf][s][0];
            float yv = 0.0f;
#pragma unroll
            for (int n = 0; n < D_STATE; ++n) {
                const float dA = __expf(dlt * An[n]);       // exp(delta*A[d,n])
                h[n] = fmaf(dA, h[n], dx * bcs[n]);
                yv   = fmaf(h[n], bcs[D_STATE + n], yv);
            }
            yptr[off] = yv + xv * Dv;
        }

        // Own async loads done, then workgroup-wide rendezvous before swap.
        wait_async0();
        __syncthreads();
        buf ^= 1;
    }
}

// ===========================================================================
// Host-side launcher (graph-capture safe: only kernel launches on `stream`).
// ===========================================================================
extern "C" void kernel_launch(void* const* d_in, const int* in_sizes, int n_in,
                              void* d_out, int out_size, void* d_ws, size_t ws_size,
                              hipStream_t stream) {
    (void)in_sizes; (void)n_in; (void)out_size; (void)ws_size;

    const float* x      = (const float*)d_in[0]; // [B,L,D]
    const float* xpw    = (const float*)d_in[1]; // [96, D]
    const float* dtw    = (const float*)d_in[2]; // [D, 64]
    const float* dtb    = (const float*)d_in[3]; // [D]
    const float* A_log  = (const float*)d_in[4]; // [D, 16]
    const float* Dvec   = (const float*)d_in[5]; // [D]
    float*       y      = (float*)d_out;         // [B,L,D]

    // Workspace: x_dbl (4096*96*4 = 1.5 MB) then delta (4096*2048*4 = 32 MB)
    float* xdbl  = (float*)d_ws;
    float* delta = (float*)((char*)d_ws + (size_t)M_TOT * K_PROJ * sizeof(float));

    ssm_xproj_wmma<<<dim3(M_TOT / 16, K_PROJ / 16), 32, 0, stream>>>(x, xpw, xdbl);
    ssm_dtproj_wmma<<<dim3(M_TOT / 16, D_INNER / 16), 32, 0, stream>>>(xdbl, dtw, dtb, delta);
    ssm_scan<<<dim3(D_INNER / 256, BATCH), 256, 0, stream>>>(x, xdbl, delta, A_log, Dvec, y);
}